// ParametricLIF_523986010150
// MI455X (gfx1250) — compile-verified
//
#include <hip/hip_runtime.h>

// Parametric LIF forward scan over time.
//   x   : [B=128, T=32, N=8192] f32
//   tau : [1] f32  -> decay = sigmoid(tau)
//   th  : [1] f32  -> threshold
//   out : [B, T, N] f32 spikes (0.0 / 1.0)
//
// Memory-bound streaming kernel: 268 MB total traffic -> ~11.5 us at 23.3 TB/s.
// Each thread owns 4 consecutive neurons (float4 column), carries the membrane
// potential u in registers, and walks T=32 steps (stride N*4 = 32 KB between
// steps). Non-temporal hints keep the one-touch stream out of L2;
// global_prefetch_b8 warms the strided next steps.

typedef float v4f __attribute__((ext_vector_type(4)));

static constexpr int B_ = 128;
static constexpr int T_ = 32;
static constexpr int N_ = 8192;
static constexpr int NV = N_ / 4;  // float4 columns per (b): 2048

__global__ __launch_bounds__(256) void
ParametricLIF_523986010150_kernel(const float* __restrict__ x,
                                  const float* __restrict__ tau,
                                  const float* __restrict__ th,
                                  float* __restrict__ out) {
    const int col = blockIdx.x * blockDim.x + threadIdx.x;  // 0 .. B_*NV-1
    const int b  = col / NV;
    const int nv = col - b * NV;

    // Uniform scalars: decay = sigmoid(tau), threshold.
    const float tval  = tau[0];
    const float decay = 1.0f / (1.0f + __expf(-tval));
    const float thv   = th[0];

    const v4f* __restrict__ xp =
        reinterpret_cast<const v4f*>(x) + (size_t)b * (size_t)(T_ * NV) + nv;
    v4f* __restrict__ op =
        reinterpret_cast<v4f*>(out) + (size_t)b * (size_t)(T_ * NV) + nv;

    v4f u = (v4f){0.0f, 0.0f, 0.0f, 0.0f};

    // Warm the first strided lines (32 KB apart in t).
    __builtin_prefetch(xp, 0, 0);
    __builtin_prefetch(xp + (size_t)1 * NV, 0, 0);

#pragma unroll
    for (int t = 0; t < T_; ++t) {
        // gfx1250 prefetch path: pull upcoming time-steps' cachelines.
        if ((t & 3) == 0 && (t + 4) < T_) {
            __builtin_prefetch(xp + (size_t)(t + 4) * NV, 0, 0);
            __builtin_prefetch(xp + (size_t)(t + 5) * NV, 0, 0);
        }

        // One-touch stream: non-temporal 128-bit load.
        const v4f xt = __builtin_nontemporal_load(xp + (size_t)t * NV);

        v4f s;
#pragma unroll
        for (int c = 0; c < 4; ++c) {
            float uc = u[c] * decay + xt[c];        // leaky integrate (v_fma_f32)
            const bool fire = (uc - thv) >= 0.0f;   // spike (v_cmp)
            s[c] = fire ? 1.0f : 0.0f;
            u[c] = fire ? 0.0f : uc;                // hard reset (v_cndmask)
        }

        // One-touch stream: non-temporal 128-bit store.
        __builtin_nontemporal_store(s, op + (size_t)t * NV);
    }
}

extern "C" void kernel_launch(void* const* d_in, const int* in_sizes, int n_in,
                              void* d_out, int out_size, void* d_ws, size_t ws_size,
                              hipStream_t stream) {
    const float* x   = (const float*)d_in[0];
    const float* tau = (const float*)d_in[1];
    const float* th  = (const float*)d_in[2];
    float* out       = (float*)d_out;

    (void)in_sizes; (void)n_in; (void)out_size; (void)d_ws; (void)ws_size;

    const int cols = B_ * NV;           // 262,144 float4 columns
    const int block = 256;              // 8 wave32 per workgroup
    const int grid = cols / block;      // 1024 workgroups

    ParametricLIF_523986010150_kernel<<<grid, block, 0, stream>>>(x, tau, th, out);
}